// GCN_7988639171255
// MI455X (gfx1250) — compile-verified
//
#include <hip/hip_runtime.h>
#include <hip/hip_bf16.h>

// ---------------------------------------------------------------------------
// GCN (2 layers, D=96) for MI455X / gfx1250, wave32.
// - GEMMs: V_WMMA_F32_16X16X4_F32, full fp32 (GEMM is not the bottleneck;
//   the 850K-edge gather/scatter dominates at ~650MB/layer -> ~28us @ 23.3TB/s,
//   largely L2-resident since h/agg are 19.2MB each << 192MB L2).
// - GEMM blocking: 256-thread block owns 64 rows; stages X stripe + full
//   W and Wr into LDS once (~102.5KB dynamic LDS, 3 blocks/WGP); 8 waves
//   compute 48 16x16 tiles (6 per wave). Row strides padded (98 / 104 floats)
//   to kill LDS bank conflicts on the WMMA operand reads.
// ---------------------------------------------------------------------------

#define D 96
#define TILE 16
#define KSTEPS (D / 4)        // 24 x wmma f32 16x16x4 over K=96
#define CTILES (D / TILE)     // 6 column tiles
#define BM 64                 // rows per GEMM block
#define MT (BM / TILE)        // 4 M-tiles per block
#define NWAVES 8
#define TPB (NWAVES * 32)     // 256 threads
#define NTILES (MT * 2 * CTILES)        // 48 output tiles per block
#define TPW (NTILES / NWAVES)           // 6 tiles per wave
#define XP 98                 // padded X row stride (floats)
#define WP 104                // padded W row stride (floats)
#define GEMM_LDS_FLOATS (BM * XP + 2 * D * WP)
#define GEMM_LDS_BYTES (GEMM_LDS_FLOATS * sizeof(float))

typedef __attribute__((ext_vector_type(2))) float v2f;
typedef __attribute__((ext_vector_type(8))) float v8f;

// ---------------------------- degree / norm --------------------------------

__global__ void deg_init_kernel(float* __restrict__ deg, int n) {
    int i = blockIdx.x * blockDim.x + threadIdx.x;
    if (i < n) deg[i] = 1.0f;   // self-loop weight
}

__global__ void deg_accum_kernel(const long long* __restrict__ col,
                                 const float* __restrict__ w,
                                 float* __restrict__ deg, int e) {
    int i = blockIdx.x * blockDim.x + threadIdx.x;
    if (i < e) atomicAdd(&deg[(int)col[i]], w[i]);
}

__global__ void deg_inv_kernel(float* __restrict__ deg, int n) {
    int i = blockIdx.x * blockDim.x + threadIdx.x;
    if (i < n) {
        float d = deg[i];
        deg[i] = (d > 0.0f) ? rsqrtf(d) : 0.0f;   // becomes dinv in-place
    }
}

// ------------------------ fused dual GEMM (WMMA) ---------------------------
// grid = ceil(n/64) blocks x 256 threads. Per block: stage X[64x96] (padded)
// and both weight matrices into LDS, then 8 waves each produce 6 of the 48
// 16x16 output tiles. Tiles jt<6: H = x@W (bias deferred); jt>=6:
// G = sigmoid(x@Wr + br).
__global__ void gemm_wmma_kernel(const float* __restrict__ X,
                                 const float* __restrict__ W,
                                 const float* __restrict__ Wr,
                                 const float* __restrict__ br,
                                 float* __restrict__ H,
                                 float* __restrict__ G,
                                 int n) {
    extern __shared__ float smem[];
    float* lds_x  = smem;                       // [BM][XP]
    float* lds_w  = lds_x + BM * XP;            // [D][WP]
    float* lds_wr = lds_w + D * WP;             // [D][WP]

    const int row_base = blockIdx.x * BM;
    const int tid = threadIdx.x;

    // Stage X stripe (coalesced global reads; padded LDS rows).
    for (int i = tid; i < BM * D; i += TPB) {
        int r = i / D, c = i % D;
        int gr = row_base + r;
        lds_x[r * XP + c] = (gr < n) ? X[(size_t)gr * D + c] : 0.0f;
    }
    // Stage both weight matrices (36KB each, L2-resident).
    for (int i = tid; i < D * D; i += TPB) {
        int r = i / D, c = i % D;
        lds_w[r * WP + c]  = W[i];
        lds_wr[r * WP + c] = Wr[i];
    }
    __syncthreads();

    const int wave = tid >> 5;
    const int lane = tid & 31;
    const int half = lane >> 4;    // 0: K={4k,4k+1}; 1: K={4k+2,4k+3}
    const int l    = lane & 15;

    for (int t = wave * TPW; t < wave * TPW + TPW; ++t) {
        const int  mt_local = t & (MT - 1);     // 0..3
        const int  jt       = t >> 2;           // 0..11
        const bool is_gate  = (jt >= CTILES);
        const int  col0     = (is_gate ? jt - CTILES : jt) * TILE;
        const float* lw     = is_gate ? lds_wr : lds_w;
        const int  arow     = mt_local * TILE + l;

        v8f acc = {0.f, 0.f, 0.f, 0.f, 0.f, 0.f, 0.f, 0.f};
        #pragma unroll
        for (int k = 0; k < KSTEPS; ++k) {
            const int kb = k * 4 + half * 2;
            v2f a, b;
            // A 16x4 (ISA 7.12.2): lanes 0-15 hold M=l, K=kb,kb+1;
            // lanes 16-31 hold M=l, K=kb,kb+1 (kb=4k+2).
            a.x = lds_x[arow * XP + kb];
            a.y = lds_x[arow * XP + kb + 1];
            // B 4x16: rows striped across lanes within a VGPR.
            b.x = lw[kb * WP + col0 + l];
            b.y = lw[(kb + 1) * WP + col0 + l];
            acc = __builtin_amdgcn_wmma_f32_16x16x4_f32(
                      /*neg_a=*/false, a, /*neg_b=*/false, b,
                      /*c_mod=*/(short)0, acc,
                      /*reuse_a=*/false, /*reuse_b=*/false);
        }

        // C/D layout: VGPR v -> M = v + 8*half, N = l.
        #pragma unroll
        for (int v = 0; v < 8; ++v) {
            int row = row_base + mt_local * TILE + v + half * 8;
            if (row < n) {
                size_t idx = (size_t)row * D + col0 + l;
                float val = acc[v];
                if (is_gate) {
                    float tt = val + br[col0 + l];
                    G[idx] = 1.0f / (1.0f + __expf(-tt));
                } else {
                    H[idx] = val;
                }
            }
        }
    }
}

// -------------------- aggregation: self-loop + scatter ---------------------

// agg[i] = dinv[i]^2 * h[i]  (self-loop norm = dinv*1*dinv); also the init.
// float4-vectorized: D/4 = 24 vectors per node.
__global__ void self_init_kernel(const float4* __restrict__ h,
                                 const float* __restrict__ dinv,
                                 float4* __restrict__ agg, int n) {
    int idx = blockIdx.x * blockDim.x + threadIdx.x;
    if (idx < n * (D / 4)) {
        float di = dinv[idx / (D / 4)];
        float s = di * di;
        float4 hv = h[idx];
        float4 o;
        o.x = s * hv.x; o.y = s * hv.y; o.z = s * hv.z; o.w = s * hv.w;
        agg[idx] = o;
    }
}

// One wave per edge: coalesced 128B gather of h[row], 3 atomic f32 adds/lane.
__global__ void scatter_kernel(const long long* __restrict__ rowi,
                               const long long* __restrict__ coli,
                               const float* __restrict__ w,
                               const float* __restrict__ dinv,
                               const float* __restrict__ h,
                               float* __restrict__ agg, int e) {
    int eidx = blockIdx.x * (blockDim.x / 32) + (threadIdx.x >> 5);
    int lane = threadIdx.x & 31;
    if (eidx >= e) return;
    int r = (int)rowi[eidx];
    int c = (int)coli[eidx];
    const float* hr = h + (size_t)r * D;
    __builtin_prefetch(hr, 0, 0);               // global_prefetch_b8
    float norm = dinv[r] * w[eidx] * dinv[c];
    float* ac = agg + (size_t)c * D;
    #pragma unroll
    for (int i = 0; i < D / 32; ++i) {
        int d = lane + i * 32;
        atomicAdd(&ac[d], norm * hr[d]);
    }
}

// -------------------------- gated residual + relu --------------------------

__global__ void finalize_kernel(const float4* __restrict__ x,
                                const float4* __restrict__ agg,
                                const float4* __restrict__ g,
                                const float4* __restrict__ b,   // D/4 vectors
                                float4* __restrict__ out, int n) {
    int idx = blockIdx.x * blockDim.x + threadIdx.x;
    if (idx < n * (D / 4)) {
        float4 bv = b[idx % (D / 4)];
        float4 xv = x[idx], av = agg[idx], gv = g[idx];
        float4 o;
        o.x = (1.0f - gv.x) * xv.x + gv.x * (av.x + bv.x);
        o.y = (1.0f - gv.y) * xv.y + gv.y * (av.y + bv.y);
        o.z = (1.0f - gv.z) * xv.z + gv.z * (av.z + bv.z);
        o.w = (1.0f - gv.w) * xv.w + gv.w * (av.w + bv.w);
        o.x = o.x > 0.f ? o.x : 0.f;
        o.y = o.y > 0.f ? o.y : 0.f;
        o.z = o.z > 0.f ? o.z : 0.f;
        o.w = o.w > 0.f ? o.w : 0.f;
        out[idx] = o;
    }
}

// -------------------------------- launch -----------------------------------

extern "C" void kernel_launch(void* const* d_in, const int* in_sizes, int n_in,
                              void* d_out, int out_size, void* d_ws, size_t ws_size,
                              hipStream_t stream) {
    const float*      x   = (const float*)d_in[0];
    const long long*  ei  = (const long long*)d_in[1];  // [2,E] int64, flat
    const float*      ew  = (const float*)d_in[2];
    const float*      W0  = (const float*)d_in[3];
    const float*      b0  = (const float*)d_in[4];
    const float*      W1  = (const float*)d_in[5];
    const float*      b1  = (const float*)d_in[6];
    const float*      Wr0 = (const float*)d_in[7];
    const float*      br0 = (const float*)d_in[8];
    const float*      Wr1 = (const float*)d_in[9];
    const float*      br1 = (const float*)d_in[10];
    float*            out = (float*)d_out;

    const int N = in_sizes[0] / D;
    const int E = in_sizes[2];
    const long long* erow = ei;       // edge_index[0] = source
    const long long* ecol = ei + E;   // edge_index[1] = target

    // Workspace layout (floats).
    float* ws   = (float*)d_ws;
    float* dinv = ws;                     // N      (deg, then dinv in-place)
    float* h    = dinv + N;               // N*D
    float* g    = h + (size_t)N * D;      // N*D
    float* agg  = g + (size_t)N * D;      // N*D
    float* x1   = agg + (size_t)N * D;    // N*D (layer-0 output)

    const int TB   = 256;
    const int gN   = (N + TB - 1) / TB;
    const int gE   = (E + TB - 1) / TB;
    const int gND4 = ((N * (D / 4)) + TB - 1) / TB;
    const int gEw  = (E + (TB / 32) - 1) / (TB / 32);     // wave per edge
    const int gGemm = (N + BM - 1) / BM;

    // Degree / normalization (layer-invariant): deg -> dinv in-place.
    deg_init_kernel<<<gN, TB, 0, stream>>>(dinv, N);
    deg_accum_kernel<<<gE, TB, 0, stream>>>(ecol, ew, dinv, E);
    deg_inv_kernel<<<gN, TB, 0, stream>>>(dinv, N);

    // ---- layer 0 ----
    gemm_wmma_kernel<<<gGemm, TPB, GEMM_LDS_BYTES, stream>>>(x, W0, Wr0, br0, h, g, N);
    self_init_kernel<<<gND4, TB, 0, stream>>>((const float4*)h, dinv, (float4*)agg, N);
    scatter_kernel<<<gEw, TB, 0, stream>>>(erow, ecol, ew, dinv, h, agg, E);
    finalize_kernel<<<gND4, TB, 0, stream>>>((const float4*)x, (const float4*)agg,
                                             (const float4*)g, (const float4*)b0,
                                             (float4*)x1, N);

    // ---- layer 1 ----
    gemm_wmma_kernel<<<gGemm, TPB, GEMM_LDS_BYTES, stream>>>(x1, W1, Wr1, br1, h, g, N);
    self_init_kernel<<<gND4, TB, 0, stream>>>((const float4*)h, dinv, (float4*)agg, N);
    scatter_kernel<<<gEw, TB, 0, stream>>>(erow, ecol, ew, dinv, h, agg, E);
    finalize_kernel<<<gND4, TB, 0, stream>>>((const float4*)x1, (const float4*)agg,
                                             (const float4*)g, (const float4*)b1,
                                             (float4*)out, N);
}